// PersistenceDiagramLayer_76390288326757
// MI455X (gfx1250) — compile-verified
//
#include <hip/hip_runtime.h>
#include <cstdint>

#define HH 28
#define WW 28
#define NN (HH * WW)   // 784 grid cells
#define MM (NN + 1)    // 785 = cells + virtual boundary vertex
#define NMAXP 100
#define NTHREADS 256
#define KMAX 8

// ---------------------------------------------------------------------------
// Union-find "find" with path halving (single-lane use during the scan).
// ---------------------------------------------------------------------------
__device__ __forceinline__ int uf_find(int* parent, int i) {
    while (parent[i] != i) {
        parent[i] = parent[parent[i]];
        i = parent[i];
    }
    return i;
}

// ---------------------------------------------------------------------------
// Neighbor enumeration, matching the reference's direction ordering exactly.
// mode 0: 8-connectivity on the 28x28 grid (K = 8)
// mode 1: 4-connectivity + virtual boundary vertex NN as 5th neighbor (K = 5)
// ---------------------------------------------------------------------------
__device__ __forceinline__ int nbr_of(int c, int k, int mode) {
    if (c >= NN) return -1;  // virtual vertex has no outgoing neighbor list
    int row = c / WW, col = c % WW;
    int dr, dc;
    if (mode == 0) {
        const int DR[8] = {-1, -1, -1, 0, 0, 1, 1, 1};
        const int DC[8] = {-1, 0, 1, -1, 1, -1, 0, 1};
        dr = DR[k]; dc = DC[k];
    } else {
        if (k == 4) {
            bool bdry = (row == 0) | (row == HH - 1) | (col == 0) | (col == WW - 1);
            return bdry ? NN : -1;
        }
        const int DR[4] = {-1, 1, 0, 0};
        const int DC[4] = {0, 0, -1, 1};
        dr = DR[k]; dc = DC[k];
    }
    int r2 = row + dr, c2 = col + dc;
    if (r2 < 0 || r2 >= HH || c2 < 0 || c2 >= WW) return -1;
    return r2 * WW + c2;
}

// ---------------------------------------------------------------------------
// Kernel: one workgroup per (sample, diagram). 64 blocks x 256 threads.
// ---------------------------------------------------------------------------
__global__ void __launch_bounds__(NTHREADS)
pd_kernel(const float* __restrict__ x, float* __restrict__ out) {
    __shared__ float s_vals[MM];
    __shared__ int   s_rank[MM];
    __shared__ int   s_order[MM];
    __shared__ int   s_parent[MM];
    __shared__ int   s_nbr[MM * KMAX];   // precomputed neighbor table
    __shared__ float s_pb[MM + 8];
    __shared__ float s_pd[MM + 8];
    __shared__ int   s_np;

    const int tid    = (int)threadIdx.x;
    const int sample = (int)blockIdx.x >> 1;
    const int mode   = (int)blockIdx.x & 1;
    const int M      = (mode == 0) ? NN : MM;
    const int K      = (mode == 0) ? 8 : 5;

    const float* xv = x + (size_t)sample * NN;
    float* outs = out + (size_t)sample * (2 * NMAXP * 2) + (size_t)mode * (NMAXP * 2);

    // Zero this block's output slice (unused slots must be zero).
    for (int i = tid; i < NMAXP * 2; i += NTHREADS) outs[i] = 0.0f;

    // CDNA5 async global->LDS copy engine: stage the 784 values directly into
    // LDS without touching VGPRs. Tracked by ASYNCcnt.
    for (int i = tid; i < NN; i += NTHREADS) {
        uint32_t lds_off = (uint32_t)(uintptr_t)(&s_vals[i]);
        const float* gp = xv + i;
        asm volatile("global_load_async_to_lds_b32 %0, %1, off"
                     :: "v"(lds_off), "v"(gp)
                     : "memory");
    }

    // Precompute neighbor table in parallel (independent of the loaded data,
    // so it overlaps the in-flight async copy).
    for (int i = tid; i < M * K; i += NTHREADS) {
        int c = i / K, k = i % K;
        s_nbr[c * KMAX + k] = nbr_of(c, k, mode);
    }

    asm volatile("s_wait_asynccnt 0x0" ::: "memory");
    __syncthreads();

    if (mode == 1) {
        // Superlevel filtration: negate values, append virtual -inf vertex.
        for (int i = tid; i < NN; i += NTHREADS) s_vals[i] = -s_vals[i];
        if (tid == 0) s_vals[NN] = -__builtin_inff();
        __syncthreads();
    }

    // ---- Parallel rank computation (O(M^2) comparisons over 8 wave32s).
    // rank[i] = #{ j : v[j] < v[i]  or (v[j] == v[i] and j < i) }  (a bijection)
    for (int i = tid; i < M; i += NTHREADS) {
        float vi = s_vals[i];
        int cnt = 0;
        for (int j = 0; j < M; ++j) {
            float vj = s_vals[j];
            if (vj < vi || (vj == vi && j < i)) cnt++;
        }
        s_rank[i] = cnt;
        s_order[cnt] = i;
        s_parent[i] = i;
    }
    __syncthreads();

    // ---- Sequential elder-rule union-find scan (lane 0; inherently serial).
    // Neighbor table is precomputed; root of c's component is tracked across
    // the k-loop (after a merge the root is `elder`), which is semantically
    // identical to re-running find(c) but removes LDS walks from the critical
    // path.
    if (tid == 0) {
        int np = 0;
        for (int r = 0; r < M; ++r) {
            int c = s_order[r];
            float vc = s_vals[c];
            const int* nrow = &s_nbr[c * KMAX];
            int rc = -1;  // lazily resolved root of c's component
            for (int k = 0; k < K; ++k) {
                int n = nrow[k];
                if (n < 0) continue;
                if (s_rank[n] >= r) continue;      // neighbor not yet in filtration
                if (rc < 0) rc = uf_find(s_parent, c);
                int rn = uf_find(s_parent, n);
                if (rc == rn) continue;
                int younger, elder;
                if (s_rank[rc] > s_rank[rn]) { younger = rc; elder = rn; }
                else                         { younger = rn; elder = rc; }
                s_parent[younger] = elder;
                float b = s_vals[younger];         // extremum of the dying component
                if (mode == 0) { s_pb[np] = b;   s_pd[np] = vc; }
                else           { s_pb[np] = -vc; s_pd[np] = -b; }  // pairs1 = (-d,-b)
                np++;
                rc = elder;                        // new root of c's component
            }
        }
        if (mode == 0) {  // essential class: (global min, +inf)
            s_pb[np] = s_vals[s_order[0]];
            s_pd[np] = __builtin_inff();
            np++;
        }
        s_np = np;
    }
    __syncthreads();

    // ---- Parallel top-NMAXP selection by persistence (rank-by-counting;
    // tie-break by lower pair index, matching lax.top_k stability).
    int np = s_np;
    for (int p = tid; p < np; p += NTHREADS) {
        float b = s_pb[p], d = s_pd[p];
        if (!(d > b)) continue;                    // pers <= 0 -> excluded
        float pers = d - b;
        int pos = 0;
        for (int q = 0; q < np; ++q) {
            float bq = s_pb[q], dq = s_pd[q];
            if (dq > bq) {
                float pq = dq - bq;
                if (pq > pers || (pq == pers && q < p)) pos++;
            }
        }
        if (pos < NMAXP) {
            outs[2 * pos + 0] = b;
            outs[2 * pos + 1] = d;
        }
    }
}

extern "C" void kernel_launch(void* const* d_in, const int* in_sizes, int n_in,
                              void* d_out, int out_size, void* d_ws, size_t ws_size,
                              hipStream_t stream) {
    (void)in_sizes; (void)n_in; (void)out_size; (void)d_ws; (void)ws_size;
    const float* x = (const float*)d_in[0];
    float* out = (float*)d_out;
    // 32 samples x 2 diagrams = 64 workgroups, 256 threads (8 wave32s) each.
    pd_kernel<<<64, NTHREADS, 0, stream>>>(x, out);
}